// DotprodAttention_3_45492293599831
// MI455X (gfx1250) — compile-verified
//
#include <hip/hip_runtime.h>

typedef __attribute__((ext_vector_type(2))) float v2f;
typedef __attribute__((ext_vector_type(4))) float v4f;
typedef __attribute__((ext_vector_type(8))) float v8f;

#define NEG_BIG (-1e30f)
#define LRELU_SLOPE 0.01f
#define EPS_BN 1e-5f

#define KB 16            // K-depth of staged B chunk
#define PAIR_STRIDE 160  // dwords per K-pair group: lanes16-31 land +32 banks

// ---------------------------------------------------------------------------
// Batched fp32 GEMM: C[z] = A[z] @ B[z] (+ bias), all row-major.
//   A: M x K (lda=K), B: K x N (ldb=N), C: M x N (ldc=N)
// Block = 128 threads (4 waves) computing a 64x64 C macro-tile:
//   wave w -> rows [row0, row0+16), all 64 cols -> 4 x v8f accumulators.
// B staged through LDS in 16x64 chunks, K-PAIR-INTERLEAVED:
//   Bs[(row>>1)*PAIR_STRIDE + col*2 + (row&1)] = B[k0+row][colBase+col]
// so each WMMA B fragment {B[k][c], B[k+1][c]} is one ds_load_b64 directly
// into the source register pair (no repacking VALU movs).
// M, N multiples of 64; K multiple of 16 (true for all call sites).
// ---------------------------------------------------------------------------
__global__ __launch_bounds__(128)
void gemm_wmma_f32(const float* __restrict__ A, const float* __restrict__ B,
                   const float* __restrict__ bias, float* __restrict__ C,
                   int M, int N, int K,
                   long long strideA, long long strideB, long long strideC)
{
    __shared__ float Bs[(KB / 2) * PAIR_STRIDE];   // 1280 floats = 5 KB

    const int lane = threadIdx.x & 31;
    const int wave = threadIdx.x >> 5;
    const int half = lane >> 4;      // 0: lanes 0-15, 1: lanes 16-31
    const int l16  = lane & 15;
    const int koff = 2 * half;       // lanes 16-31 own K+2,K+3 (ISA A/B layout)

    const int colBase = blockIdx.x * 64;
    const int row0    = blockIdx.y * 64 + wave * 16;

    const float* Ab = A + (long long)blockIdx.z * strideA;
    const float* Bb = B + (long long)blockIdx.z * strideB;
    float*       Cb = C + (long long)blockIdx.z * strideC;

    const float* arow = Ab + (size_t)(row0 + l16) * K;

    // staging coordinates: thread t moves 8 consecutive floats of one B row
    const int sr   = threadIdx.x >> 3;        // 0..15  (row within chunk)
    const int sc   = (threadIdx.x & 7) * 8;   // 0,8,...,56 (col within panel)
    const int sbase = (sr >> 1) * PAIR_STRIDE + (sr & 1);  // interleaved base

    v8f acc[4] = {};

    for (int k0 = 0; k0 < K; k0 += KB) {
        __syncthreads();   // previous chunk fully consumed
        {
            const float* src = Bb + (size_t)(k0 + sr) * N + colBase + sc;
            v4f x0 = *(const v4f*)(src);
            v4f x1 = *(const v4f*)(src + 4);
            #pragma unroll
            for (int i = 0; i < 4; ++i) Bs[sbase + (sc + i) * 2]     = x0[i];
            #pragma unroll
            for (int i = 0; i < 4; ++i) Bs[sbase + (sc + 4 + i) * 2] = x1[i];
            if (k0 + KB < K)   // prefetch next chunk (-> global_prefetch_b8)
                __builtin_prefetch(src + (size_t)KB * N, 0, 0);
        }
        __syncthreads();   // chunk visible to all waves

        #pragma unroll
        for (int kk = 0; kk < KB; kk += 4) {
            v2f a = *(const v2f*)(arow + k0 + kk + koff);
            const int kgrp = (kk + koff) >> 1;   // even: exact pair group
            #pragma unroll
            for (int ct = 0; ct < 4; ++ct) {
                // one ds_load_b64: {B[k][col], B[k+1][col]} register-adjacent
                v2f b = *(const v2f*)(&Bs[kgrp * PAIR_STRIDE + (ct * 16 + l16) * 2]);
                acc[ct] = __builtin_amdgcn_wmma_f32_16x16x4_f32(
                              false, a, false, b, (short)0, acc[ct], false, false);
            }
        }
    }

    if (bias != nullptr) {
        #pragma unroll
        for (int ct = 0; ct < 4; ++ct) {
            float bb = bias[colBase + ct * 16 + l16];
            #pragma unroll
            for (int r = 0; r < 8; ++r) acc[ct][r] += bb;
        }
    }

    #pragma unroll
    for (int ct = 0; ct < 4; ++ct) {
        #pragma unroll
        for (int r = 0; r < 8; ++r)
            Cb[(size_t)(row0 + r + 8 * half) * N + colBase + ct * 16 + l16] = acc[ct][r];
    }
}

// ---------------------------------------------------------------------------
// Per-batch attention scores + masking + GLOBAL softmax over all N*N logits.
// One 256-thread workgroup per batch (B = 8, N = 128, H = 64).
// ---------------------------------------------------------------------------
__global__ __launch_bounds__(256)
void gat_scores_softmax(const float* __restrict__ s_src, const float* __restrict__ s_tgt,
                        const int*   __restrict__ adj,   const float* __restrict__ ab1,
                        const float* __restrict__ A2,    const float* __restrict__ ab2,
                        float* __restrict__ attn)
{
    __shared__ float sb1[64];
    __shared__ float sa2[64];
    __shared__ float red[256];
    __shared__ float bcast;

    const int b   = blockIdx.x;
    const int tid = threadIdx.x;
    if (tid < 64) { sb1[tid] = ab1[tid]; sa2[tid] = A2[tid]; }
    __syncthreads();

    const float bias2 = ab2[0];
    const int NN = 128 * 128;
    const float* Ss = s_src + (size_t)b * 128 * 64;
    const float* St = s_tgt + (size_t)b * 128 * 64;
    const int*   Am = adj   + (size_t)b * NN;
    float*       Eb = attn  + (size_t)b * NN;

    float lmax = -3.4e38f;
    for (int idx = tid; idx < NN; idx += 256) {
        const int i = idx >> 7, j = idx & 127;
        const float* si = Ss + i * 64;
        const float* sj = St + j * 64;
        float acc = bias2;
        #pragma unroll 8
        for (int hh = 0; hh < 64; ++hh) {
            float v = si[hh] + sj[hh] + sb1[hh];
            acc += (v > 0.f ? v : 0.f) * sa2[hh];
        }
        float e = (acc > 0.f) ? acc : LRELU_SLOPE * acc;   // leaky relu
        float m = (float)Am[idx];
        float logit = e * m + (1.f - m) * NEG_BIG;          // mask
        Eb[idx] = logit;
        lmax = fmaxf(lmax, logit);
    }
    red[tid] = lmax; __syncthreads();
    for (int s = 128; s > 0; s >>= 1) {
        if (tid < s) red[tid] = fmaxf(red[tid], red[tid + s]);
        __syncthreads();
    }
    if (tid == 0) bcast = red[0];
    __syncthreads();
    const float gmax = bcast;

    float lsum = 0.f;
    for (int idx = tid; idx < NN; idx += 256) lsum += __expf(Eb[idx] - gmax);
    __syncthreads();
    red[tid] = lsum; __syncthreads();
    for (int s = 128; s > 0; s >>= 1) {
        if (tid < s) red[tid] += red[tid + s];
        __syncthreads();
    }
    if (tid == 0) bcast = red[0];
    __syncthreads();
    const float inv = 1.f / bcast;

    for (int idx = tid; idx < NN; idx += 256)
        Eb[idx] = __expf(Eb[idx] - gmax) * inv;
}

// ---------------------------------------------------------------------------
// g[b,d] = sum_i node[b,i,d]; batch-norm over b (B=8); one thread per d.
// ---------------------------------------------------------------------------
__global__ __launch_bounds__(256)
void graph_norm(const float* __restrict__ node, const float* __restrict__ gamma,
                const float* __restrict__ beta, float* __restrict__ out)
{
    int d = blockIdx.x * blockDim.x + threadIdx.x;
    if (d >= 768) return;
    float g[8];
    float mean = 0.f;
    #pragma unroll
    for (int b = 0; b < 8; ++b) {
        const float* p = node + (size_t)b * 128 * 768 + d;
        float s = 0.f;
        for (int i = 0; i < 128; ++i) s += p[(size_t)i * 768];
        g[b] = s; mean += s;
    }
    mean *= 0.125f;
    float var = 0.f;
    #pragma unroll
    for (int b = 0; b < 8; ++b) { float t = g[b] - mean; var += t * t; }
    var *= 0.125f;
    const float inv = rsqrtf(var + EPS_BN);
    const float ga = gamma[d], be = beta[d];
    #pragma unroll
    for (int b = 0; b < 8; ++b)
        out[b * 768 + d] = ga * (g[b] - mean) * inv + be;
}

// ---------------------------------------------------------------------------
// Orchestration. Input order (setup_inputs): feature, aspect(unused), adj,
// W0, b0, W1, b1, A1, ab1, A2, ab2, gamma, beta.
// d_out = [graph_feature (8*768) | node_feature (8*128*768)].
// ---------------------------------------------------------------------------
extern "C" void kernel_launch(void* const* d_in, const int* in_sizes, int n_in,
                              void* d_out, int out_size, void* d_ws, size_t ws_size,
                              hipStream_t stream)
{
    (void)in_sizes; (void)n_in; (void)out_size; (void)ws_size;

    const float* feature = (const float*)d_in[0];
    const int*   adj     = (const int*)  d_in[2];
    const float* W0      = (const float*)d_in[3];
    const float* b0      = (const float*)d_in[4];
    const float* W1      = (const float*)d_in[5];
    const float* b1      = (const float*)d_in[6];
    const float* A1      = (const float*)d_in[7];
    const float* ab1     = (const float*)d_in[8];
    const float* A2      = (const float*)d_in[9];
    const float* ab2     = (const float*)d_in[10];
    const float* gamma   = (const float*)d_in[11];
    const float* beta    = (const float*)d_in[12];

    float* ws    = (float*)d_ws;
    float* h     = ws;                       // 1024*768
    float* ssrc  = h    + 1024 * 768;        // 1024*64
    float* stgt  = ssrc + 1024 * 64;         // 1024*64
    float* attn  = stgt + 1024 * 64;         // 8*128*128
    float* feat1 = attn + 8 * 128 * 128;     // 1024*768 (layer-1 output)

    float* graph_out = (float*)d_out;        // 8*768
    float* node_out  = graph_out + 8 * 768;  // 8*128*768

    const dim3 blk(128);
    const float* fin = feature;
    const float* W   = W0;
    const float* bb  = b0;
    float*       fout = feat1;

    for (int layer = 0; layer < 2; ++layer) {
        // h = fin @ W + b : M=1024, N=768, K=768
        gemm_wmma_f32<<<dim3(768 / 64, 1024 / 64, 1), blk, 0, stream>>>(
            fin, W, bb, h, 1024, 768, 768, 0, 0, 0);
        // s_src = h @ A1[:768], s_tgt = h @ A1[768:] : M=1024, N=64, K=768
        gemm_wmma_f32<<<dim3(1, 1024 / 64, 1), blk, 0, stream>>>(
            h, A1,            nullptr, ssrc, 1024, 64, 768, 0, 0, 0);
        gemm_wmma_f32<<<dim3(1, 1024 / 64, 1), blk, 0, stream>>>(
            h, A1 + 768 * 64, nullptr, stgt, 1024, 64, 768, 0, 0, 0);
        // scores + mask + global softmax (one WG per batch)
        gat_scores_softmax<<<dim3(8), dim3(256), 0, stream>>>(
            ssrc, stgt, adj, ab1, A2, ab2, attn);
        // fout[b] = attn[b](128x128) @ h[b](128x768), batched over z
        gemm_wmma_f32<<<dim3(768 / 64, 128 / 64, 8), blk, 0, stream>>>(
            attn, h, nullptr, fout, 128, 768, 128,
            128LL * 128, 128LL * 768, 128LL * 768);

        fin = fout; W = W1; bb = b1; fout = node_out;
    }

    // graph_feature from node_feature (already in d_out)
    graph_norm<<<dim3(3), dim3(256), 0, stream>>>(node_out, gamma, beta, graph_out);
}